// GroupDynamicDWConv_51496657879116
// MI455X (gfx1250) — compile-verified
//
#include <hip/hip_runtime.h>
#include <hip/hip_bf16.h>

#define BATCH 16
#define DIM   256
#define KW    7
#define G     4
#define CPG   64
#define HID   64
#define GKK   196   // G*K*K
#define HW    64
#define PLANE (HW*HW)      // 4096
#define HALO  70           // 64 + 6
#define HALOSZ (HALO*HALO) // 4900

typedef __attribute__((ext_vector_type(2))) float v2f;
typedef __attribute__((ext_vector_type(8))) float v8f;

// ---------------------------------------------------------------------------
// Kernel 1: global average pool over each (b,c) 64x64 plane -> pooled[b*256+c]
// ---------------------------------------------------------------------------
__global__ __launch_bounds__(256) void pool_kernel(const float* __restrict__ x,
                                                   float* __restrict__ pooled) {
    const int bc = blockIdx.x;
    const float4* xp = (const float4*)(x + (size_t)bc * PLANE);
    const int tid = threadIdx.x;
    float s = 0.0f;
    #pragma unroll
    for (int i = 0; i < 4; ++i) {               // 1024 float4 / 256 threads
        float4 v = xp[tid + i * 256];
        s += v.x + v.y + v.z + v.w;
    }
    __shared__ float red[256];
    red[tid] = s;
    __syncthreads();
    #pragma unroll
    for (int off = 128; off > 0; off >>= 1) {
        if (tid < off) red[tid] += red[tid + off];
        __syncthreads();
    }
    if (tid == 0) pooled[bc] = red[0] * (1.0f / (float)PLANE);
}

// ---------------------------------------------------------------------------
// Kernel 2: both squeeze MLPs via V_WMMA_F32_16X16X4_F32 (full fp32 GEMMs).
//   dyn[b, g*49+t]  = pooled @ dk_w1^T -> GELU -> @ dk_w2^T + b2
//   gate[b, c]      = sigmoid(GELU(pooled @ cg_w1^T + b1) @ cg_w2^T + b2)
// One block, 256 threads = 8 waves; each wave owns 16x16 output tiles.
// A layout (16x4 f32): lanes 0-15 hold M=lane, K=k0/k0+1 in vgpr0/1;
//                      lanes 16-31 hold M=lane-16, K=k0+2/k0+3.
// B layout (4x16 f32): vgpr v, lanes 0-15 = row K=k0+v, lanes 16-31 = K=k0+v+2.
// C/D layout: vgpr v, lanes 0-15 -> (M=v, N=lane); lanes 16-31 -> (M=v+8).
// ---------------------------------------------------------------------------
__device__ __forceinline__ float gelu_exact(float v) {
    return 0.5f * v * (1.0f + erff(v * 0.70710678118654752f));
}

__global__ __launch_bounds__(256) void mlp_kernel(
    const float* __restrict__ pooled,
    const float* __restrict__ dk_w1, const float* __restrict__ dk_b1,
    const float* __restrict__ dk_w2, const float* __restrict__ dk_b2,
    const float* __restrict__ cg_w1, const float* __restrict__ cg_b1,
    const float* __restrict__ cg_w2, const float* __restrict__ cg_b2,
    float* __restrict__ dyn, float* __restrict__ gate)
{
    __shared__ float pool_s[BATCH * DIM];       // 16 KB
    __shared__ float h_s[2][BATCH * HID];       // 8 KB (dk / cg hidden acts)

    const int tid  = threadIdx.x;
    const int wave = tid >> 5;
    const int lane = tid & 31;
    const int hl   = lane >> 4;                 // lane half: 0 or 1
    const int lm   = lane & 15;

    for (int i = tid; i < BATCH * DIM; i += 256) pool_s[i] = pooled[i];
    __syncthreads();

    // ---- phase 1: h = GELU(pooled @ W1^T + b1), both branches, 8 tiles ----
    {
        const int branch = wave >> 2;           // 0 = dk, 1 = cg
        const int tn     = wave & 3;            // N tile within HID=64
        const float* w1 = branch ? cg_w1 : dk_w1;   // (HID, DIM) row-major
        const float* b1 = branch ? cg_b1 : dk_b1;
        const int n = tn * 16 + lm;             // output column (0..63)
        v8f acc = {};
        for (int k0 = 0; k0 < DIM; k0 += 4) {
            v2f a, b;
            a.x = pool_s[lm * DIM + k0 + hl * 2];
            a.y = pool_s[lm * DIM + k0 + hl * 2 + 1];
            b.x = w1[n * DIM + k0 + hl * 2];        // B[k][n] = w1[n][k]
            b.y = w1[n * DIM + k0 + hl * 2 + 1];
            acc = __builtin_amdgcn_wmma_f32_16x16x4_f32(
                false, a, false, b, (short)0, acc, false, false);
        }
        const float bias = b1[n];
        #pragma unroll
        for (int v = 0; v < 8; ++v) {
            const int m = v + hl * 8;
            h_s[branch][m * HID + n] = gelu_exact(acc[v] + bias);
        }
    }
    __syncthreads();

    // ---- phase 2: 13 dyn tiles (N=196) + 16 gate tiles (N=256) = 29 -------
    for (int t = wave; t < 29; t += 8) {        // wave-uniform loop
        const int isGate = (t >= 13) ? 1 : 0;
        const int tn     = isGate ? (t - 13) : t;
        const float* hs = h_s[isGate];
        const float* w2 = isGate ? cg_w2 : dk_w2;   // (N, HID) row-major
        const int N = isGate ? DIM : GKK;
        const int n = tn * 16 + lm;
        v8f acc = {};
        for (int k0 = 0; k0 < HID; k0 += 4) {
            v2f a, b;
            a.x = hs[lm * HID + k0 + hl * 2];
            a.y = hs[lm * HID + k0 + hl * 2 + 1];
            float bx = 0.0f, by = 0.0f;
            if (n < N) {                        // only lane-diverges the loads
                bx = w2[n * HID + k0 + hl * 2];
                by = w2[n * HID + k0 + hl * 2 + 1];
            }
            b.x = bx; b.y = by;
            acc = __builtin_amdgcn_wmma_f32_16x16x4_f32(
                false, a, false, b, (short)0, acc, false, false);
        }
        if (n < N) {
            if (isGate) {
                const float bias = cg_b2[n];
                #pragma unroll
                for (int v = 0; v < 8; ++v) {
                    const int m = v + hl * 8;
                    const float s = acc[v] + bias;
                    gate[m * DIM + n] = 1.0f / (1.0f + __expf(-s));
                }
            } else {
                const float bias = dk_b2[n];
                #pragma unroll
                for (int v = 0; v < 8; ++v) {
                    const int m = v + hl * 8;
                    dyn[m * GKK + n] = acc[v] + bias;
                }
            }
        }
    }
}

// ---------------------------------------------------------------------------
// Kernel 3: per-(b,c) depthwise 7x7 conv + gate.  One block per plane.
// Fused weights w = dw[c] + dyn[b, c/64] recomputed from uniform (SGPR)
// addresses; 70x70 halo tile in LDS; each thread computes a 1x16 column
// strip (154 conflict-free LDS reads for 784 FMAs).
// ---------------------------------------------------------------------------
__global__ __launch_bounds__(256) void conv_kernel(
    const float* __restrict__ x,
    const float* __restrict__ dw,       // (256, 49)
    const float* __restrict__ dyn,      // (16, 196)
    const float* __restrict__ gate,     // (16, 256)
    float* __restrict__ out)
{
    const int bc = blockIdx.x;          // b*256 + c
    const int c  = bc & (DIM - 1);
    const int b  = bc >> 8;
    const int g  = c >> 6;

    __shared__ float tile[HALOSZ];      // 19600 B
    const float* xp = x + (size_t)bc * PLANE;
    const int tid = threadIdx.x;

    // Load halo tile with zero padding (halo coord = global + 3).
    for (int idx = tid; idx < HALOSZ; idx += 256) {
        const int ty = idx / HALO;
        const int tc = idx - ty * HALO;
        const int gy = ty - 3, gx = tc - 3;
        float v = 0.0f;
        if ((unsigned)gy < (unsigned)HW && (unsigned)gx < (unsigned)HW)
            v = xp[gy * HW + gx];
        tile[idx] = v;
    }

    // Fused per-plane weights: uniform addresses -> scalar loads.
    const float* dwp  = dw  + c * 49;
    const float* dynp = dyn + b * GKK + g * 49;
    float w[49];
    #pragma unroll
    for (int t = 0; t < 49; ++t) w[t] = dwp[t] + dynp[t];
    const float gt = gate[bc];

    __syncthreads();

    const int tx  = tid & 63;           // output column 0..63
    const int y0  = (tid >> 6) * 16;    // output row base: 0,16,32,48

    float acc[16];
    #pragma unroll
    for (int i = 0; i < 16; ++i) acc[i] = 0.0f;

    #pragma unroll
    for (int r = 0; r < 22; ++r) {      // halo rows y0 .. y0+21
        float row[7];
        #pragma unroll
        for (int q = 0; q < 7; ++q)
            row[q] = tile[(y0 + r) * HALO + tx + q];
        #pragma unroll
        for (int i = 0; i < 16; ++i) {
            const int ky = r - i;       // compile-time after unroll
            if (ky >= 0 && ky < 7) {
                #pragma unroll
                for (int kx = 0; kx < 7; ++kx)
                    acc[i] = fmaf(w[ky * 7 + kx], row[kx], acc[i]);
            }
        }
    }

    float* op = out + (size_t)bc * PLANE + tx;
    #pragma unroll
    for (int i = 0; i < 16; ++i)
        op[(y0 + i) * HW] = acc[i] * gt;
}

// ---------------------------------------------------------------------------
extern "C" void kernel_launch(void* const* d_in, const int* in_sizes, int n_in,
                              void* d_out, int out_size, void* d_ws, size_t ws_size,
                              hipStream_t stream) {
    const float* x     = (const float*)d_in[0];
    const float* dw    = (const float*)d_in[1];
    const float* dk_w1 = (const float*)d_in[2];
    const float* dk_b1 = (const float*)d_in[3];
    const float* dk_w2 = (const float*)d_in[4];
    const float* dk_b2 = (const float*)d_in[5];
    const float* cg_w1 = (const float*)d_in[6];
    const float* cg_b1 = (const float*)d_in[7];
    const float* cg_w2 = (const float*)d_in[8];
    const float* cg_b2 = (const float*)d_in[9];
    float* out = (float*)d_out;

    float* pooled = (float*)d_ws;                 // 16*256      = 4096 f
    float* dyn    = pooled + BATCH * DIM;         // 16*196      = 3136 f
    float* gate   = dyn    + BATCH * GKK;         // 16*256      = 4096 f

    pool_kernel<<<BATCH * DIM, 256, 0, stream>>>(x, pooled);
    mlp_kernel<<<1, 256, 0, stream>>>(pooled,
                                      dk_w1, dk_b1, dk_w2, dk_b2,
                                      cg_w1, cg_b1, cg_w2, cg_b2,
                                      dyn, gate);
    conv_kernel<<<BATCH * DIM, 256, 0, stream>>>(x, dw, dyn, gate, out);
}